// YoloLoss_52364241272835
// MI455X (gfx1250) — compile-verified
//
#include <hip/hip_runtime.h>

// ---------------------------------------------------------------------------
// YOLO loss on MI455X (gfx1250). Bandwidth-bound: ~354 MB read once -> scalar.
//   * Staging:  GLOBAL_LOAD_ASYNC_TO_LDS_B128 (ASYNCcnt path) when available,
//               else coalesced non-temporal b128 loads + ds_store.
//   * Compute:  one cell per thread from LDS (conflict-free strides 30/25).
//   * Reduce:   deterministic WMMA tree (v_wmma_f32_16x16x4_f32, all-ones B).
// ---------------------------------------------------------------------------

#if defined(__AMDGCN__) &&                                            \
    __has_builtin(__builtin_amdgcn_global_load_async_to_lds_b128) &&  \
    __has_builtin(__builtin_amdgcn_s_wait_asynccnt)
#define YOLO_ASYNC_LDS 1
#endif

namespace {

constexpr int   kS     = 7;
constexpr int   kPCH   = 30;   // C + 5*B channels in predictions
constexpr int   kTCH   = 25;   // C + 5 channels in target
constexpr int   kTPB   = 256;  // threads per block == cells per block
constexpr float kEps   = 1e-6f;
constexpr float kInvS  = 1.0f / 7.0f;

typedef float v2f __attribute__((ext_vector_type(2)));
typedef float v4f __attribute__((ext_vector_type(4)));   // raw vector (not HIP class)
typedef float v8f __attribute__((ext_vector_type(8)));

#ifdef YOLO_ASYNC_LDS
// Signature (from clang diagnostic): (int4 AS(1)*, int4 AS(3)*, imm, imm)
typedef int v4i __attribute__((ext_vector_type(4)));
typedef __attribute__((address_space(1))) v4i* gasync_t;
typedef __attribute__((address_space(3))) v4i* lasync_t;
#endif

__device__ __forceinline__ void to_corners(float x, float y, float w, float h,
                                           float colf, float rowf,
                                           float& x1, float& y1, float& x2, float& y2) {
  float cx = x * kInvS + colf;
  float cy = y * kInvS + rowf;
  float hw = 0.5f * w;
  float hh = 0.5f * h;
  x1 = cx - hw; y1 = cy - hh; x2 = cx + hw; y2 = cy + hh;
}

__device__ __forceinline__ float iou_corners(float ax1, float ay1, float ax2, float ay2,
                                             float bx1, float by1, float bx2, float by2) {
  float x1 = fmaxf(ax1, bx1);
  float y1 = fmaxf(ay1, by1);
  float x2 = fminf(ax2, bx2);
  float y2 = fminf(ay2, by2);
  float inter  = fmaxf(x2 - x1, 0.0f) * fmaxf(y2 - y1, 0.0f);
  float area_a = fabsf((ax2 - ax1) * (ay2 - ay1));
  float area_b = fabsf((bx2 - bx1) * (by2 - by1));
  return inter / (area_a + area_b - inter + kEps);
}

__device__ __forceinline__ float sgnf(float x) {
  return (x > 0.0f) ? 1.0f : ((x < 0.0f) ? -1.0f : 0.0f);
}

// Sum 64 wave-resident floats per iteration into an f32 16x16 accumulator
// using v_wmma_f32_16x16x4_f32 with an all-ones B matrix.
__device__ __forceinline__ v8f wmma_accum64(v8f acc, v2f a) {
  v2f ones;
  ones[0] = 1.0f;
  ones[1] = 1.0f;
  return __builtin_amdgcn_wmma_f32_16x16x4_f32(
      /*neg_a=*/false, a, /*neg_b=*/false, ones,
      /*c_mod=*/(short)0, acc, /*reuse_a=*/false, /*reuse_b=*/false);
}

// Collapse the 16x16 accumulator (rows replicated over columns) to a scalar.
// acc[r] on lanes 0-15 holds row M=r, lanes 16-31 hold row M=r+8.
__device__ __forceinline__ float wmma_collapse(v8f acc) {
  float s = acc[0] + acc[1] + acc[2] + acc[3] + acc[4] + acc[5] + acc[6] + acc[7];
  return __shfl(s, 0, 32) + __shfl(s, 16, 32);
}

} // namespace

// ---------------------------------------------------------------------------
// Kernel 1: per-block staged load + per-cell loss + WMMA block reduction.
// Assumes ncells % 256 == 0 (true: 32768*49 = 6272*256).
// ---------------------------------------------------------------------------
__global__ __launch_bounds__(kTPB) void yolo_loss_main(
    const float* __restrict__ pred, const float* __restrict__ tgt,
    float* __restrict__ partials) {
  __shared__ __align__(16) float sp[kTPB * kPCH];  // 7680 f  (30720 B)
  __shared__ __align__(16) float st[kTPB * kTCH];  // 6400 f  (25600 B)
  __shared__ float red[kTPB];

  const int tid = threadIdx.x;
  const int blk = blockIdx.x;

#ifdef YOLO_ASYNC_LDS
  // ---- Stage via async engine: memory -> LDS directly, no VGPR round-trip --
  {
    const float* gp = pred + (size_t)blk * (kTPB * kPCH);
    for (int i = tid; i < (kTPB * kPCH) / 4; i += kTPB)
      __builtin_amdgcn_global_load_async_to_lds_b128(
          (gasync_t)(gp + 4 * i), (lasync_t)(sp + 4 * i), 0, 0);

    const float* gt = tgt + (size_t)blk * (kTPB * kTCH);
    for (int i = tid; i < (kTPB * kTCH) / 4; i += kTPB)
      __builtin_amdgcn_global_load_async_to_lds_b128(
          (gasync_t)(gt + 4 * i), (lasync_t)(st + 4 * i), 0, 0);

    __builtin_amdgcn_s_wait_asynccnt(0);
  }
#else
  // ---- Fallback: coalesced non-temporal b128 loads + ds stores -------------
  {
    const v4f* gp = (const v4f*)(pred + (size_t)blk * (kTPB * kPCH));
    v4f* lp = (v4f*)sp;
    #pragma unroll
    for (int i = tid; i < (kTPB * kPCH) / 4; i += kTPB)
      lp[i] = __builtin_nontemporal_load(&gp[i]);

    const v4f* gt = (const v4f*)(tgt + (size_t)blk * (kTPB * kTCH));
    v4f* lt = (v4f*)st;
    #pragma unroll
    for (int i = tid; i < (kTPB * kTCH) / 4; i += kTPB)
      lt[i] = __builtin_nontemporal_load(&gt[i]);
  }
#endif
  __syncthreads();

  // ---- Per-cell loss --------------------------------------------------------
  float p[kPCH];
  #pragma unroll
  for (int c = 0; c < kPCH; ++c) p[c] = sp[tid * kPCH + c];
  float t[kTCH];
  #pragma unroll
  for (int c = 0; c < kTCH; ++c) t[c] = st[tid * kTCH + c];

  const int cell = blk * kTPB + tid;
  const int sc   = cell % (kS * kS);
  const float rowf = (float)(sc / kS) * kInvS;
  const float colf = (float)(sc % kS) * kInvS;

  float ax1, ay1, ax2, ay2;  // pred box 1
  float bx1, by1, bx2, by2;  // pred box 2
  float tx1, ty1, tx2, ty2;  // target box
  to_corners(p[0], p[1], p[2], p[3], colf, rowf, ax1, ay1, ax2, ay2);
  to_corners(p[5], p[6], p[7], p[8], colf, rowf, bx1, by1, bx2, by2);
  to_corners(t[0], t[1], t[2], t[3], colf, rowf, tx1, ty1, tx2, ty2);

  const float iou1 = iou_corners(ax1, ay1, ax2, ay2, tx1, ty1, tx2, ty2);
  const float iou2 = iou_corners(bx1, by1, bx2, by2, tx1, ty1, tx2, ty2);
  const float best = (iou2 > iou1) ? 1.0f : 0.0f;
  const float e    = t[4];
  const float ne   = 1.0f - e;

  // box xy + wh losses
  float bp0 = e * (best * p[5] + (1.0f - best) * p[0]);
  float bp1 = e * (best * p[6] + (1.0f - best) * p[1]);
  float bp2 = e * (best * p[7] + (1.0f - best) * p[2]);
  float bp3 = e * (best * p[8] + (1.0f - best) * p[3]);
  float bt0 = e * t[0], bt1 = e * t[1], bt2 = e * t[2], bt3 = e * t[3];

  float dxy0 = bp0 - bt0, dxy1 = bp1 - bt1;
  float box_loss_xy = dxy0 * dxy0 + dxy1 * dxy1;

  float whp2 = sgnf(bp2) * sqrtf(fabsf(bp2) + kEps);
  float whp3 = sgnf(bp3) * sqrtf(fabsf(bp3) + kEps);
  float wht2 = sqrtf(bt2);
  float wht3 = sqrtf(bt3);
  float dwh2 = whp2 - wht2, dwh3 = whp3 - wht3;
  float box_loss_wh = dwh2 * dwh2 + dwh3 * dwh3;

  // objectness
  float pred_box = best * p[9] + (1.0f - best) * p[4];
  float dobj = e * pred_box - e * t[4];
  float object_loss = dobj * dobj;

  float dn0 = ne * p[4] - ne * t[4];
  float dn1 = ne * p[9] - ne * t[4];
  float no_object_loss = dn0 * dn0 + dn1 * dn1;

  // class loss
  float class_loss = 0.0f;
  #pragma unroll
  for (int c = 0; c < 20; ++c) {
    float d = e * p[10 + c] - e * t[5 + c];
    class_loss += d * d;
  }

  float lsum = 5.0f * (box_loss_xy + box_loss_wh) + object_loss +
               0.5f * no_object_loss + class_loss;

  // ---- Block reduction: wave 0 folds 256 partials with WMMA ---------------
  red[tid] = lsum;
  __syncthreads();

  if (tid < 32) {  // full wave32 -> EXEC all-ones, legal for WMMA
    v8f acc = {};
    #pragma unroll
    for (int i = 0; i < kTPB; i += 64) {
      v2f a;
      a[0] = red[i + 2 * tid];
      a[1] = red[i + 2 * tid + 1];
      acc = wmma_accum64(acc, a);
    }
    float tot = wmma_collapse(acc);
    if (tid == 0) partials[blk] = tot;
  }
}

// ---------------------------------------------------------------------------
// Kernel 2: single-wave WMMA reduction of block partials -> final scalar.
// ---------------------------------------------------------------------------
__global__ __launch_bounds__(32) void yolo_loss_reduce(
    const float* __restrict__ partials, int n, float* __restrict__ out) {
  const int lane = threadIdx.x;
  v8f acc = {};
  for (int i = 0; i < n; i += 64) {
    int idx = i + 2 * lane;
    v2f a;
    a[0] = (idx     < n) ? partials[idx]     : 0.0f;
    a[1] = (idx + 1 < n) ? partials[idx + 1] : 0.0f;
    acc = wmma_accum64(acc, a);
  }
  float tot = wmma_collapse(acc);
  if (lane == 0) out[0] = tot;
}

// ---------------------------------------------------------------------------
extern "C" void kernel_launch(void* const* d_in, const int* in_sizes, int n_in,
                              void* d_out, int out_size, void* d_ws, size_t ws_size,
                              hipStream_t stream) {
  const float* pred = (const float*)d_in[0];   // (32768, 7, 7, 30) f32
  const float* tgt  = (const float*)d_in[1];   // (32768, 7, 7, 25) f32
  float* out        = (float*)d_out;           // scalar f32
  float* partials   = (float*)d_ws;            // one f32 per block

  const int ncells  = in_sizes[1] / kTCH;              // 1,605,632
  const int nblocks = (ncells + kTPB - 1) / kTPB;      // 6,272 (exact)

  yolo_loss_main<<<nblocks, kTPB, 0, stream>>>(pred, tgt, partials);
  yolo_loss_reduce<<<1, 32, 0, stream>>>(partials, nblocks, out);
}